// FromReference_86766929314319
// MI455X (gfx1250) — compile-verified
//
#include <hip/hip_runtime.h>

typedef __attribute__((ext_vector_type(2))) float v2f;
typedef __attribute__((ext_vector_type(8))) float v8f;

#define NI   64      // B*S images
#define RSQ  16384   // 128*128

// workspace layout (float offsets)
#define OFF_BA  0u          // 128*32  : forward-Y basis (cos | -sin)
#define OFF_BB  4096u       // 32*128 er, 32*128 ei : forward-X basis
#define OFF_F   12288u      // 128*32 fr, 128*32 fi : inverse-X basis (+sin)
#define OFF_E   20480u      // 16*128 ec, 16*128 es : inverse-Y basis (scaled)
#define OFF_XFR 24576u
#define OFF_XFI (OFF_XFR + 2097152u)
#define OFF_CMR (OFF_XFI + 2097152u)
#define OFF_CMI (OFF_CMR + 2097152u)
#define OFF_HR  (OFF_CMI + 2097152u)
#define OFF_HI  (OFF_HR  + 8388608u)
// total = OFF_HI + 8388608 = 25,190,400 floats ~ 96.1 MB of d_ws

__device__ inline v8f wmma4(v2f a, v2f b, v8f c) {
  // D = A(16x4) * B(4x16) + C, fp32
  return __builtin_amdgcn_wmma_f32_16x16x4_f32(false, a, false, b, (short)0, c,
                                               false, false);
}

// ---------------- basis init (exact twiddles, phase reduced mod 128) --------
__global__ void fno_init_basis(float* __restrict__ ws) {
  int i = blockIdx.x * blockDim.x + threadIdx.x;
  const float w0 = 6.28318530717958647692f / 128.0f;
  if (i < 4096) {  // BA[y*32+k]: k<16 -> cos(2pi k y/128), else -sin
    int y = i >> 5, k = i & 31, kk = k & 15;
    int p = (kk * y) & 127;
    float s, c; __sincosf(w0 * (float)p, &s, &c);
    ws[OFF_BA + i] = (k < 16) ? c : -s;
  }
  if (i < 4096) {  // BB: er[j*128+X]=cos, ei=-sin, kx' = j<16? j : 96+j
    int j = i >> 7, X = i & 127;
    int kx = (j < 16) ? j : (96 + j);
    int p = (kx * X) & 127;
    float s, c; __sincosf(w0 * (float)p, &s, &c);
    ws[OFF_BB + i] = c;
    ws[OFF_BB + 4096 + i] = -s;
  }
  if (i < 4096) {  // F: fr[X*32+j]=cos, fi=+sin (inverse transform)
    int X = i >> 5, j = i & 31;
    int kx = (j < 16) ? j : (96 + j);
    int p = (kx * X) & 127;
    float s, c; __sincosf(w0 * (float)p, &s, &c);
    ws[OFF_F + i] = c;
    ws[OFF_F + 4096 + i] = s;
  }
  if (i < 2048) {  // E: ec[ky*128+Y], es[...]; scale s_ky/R^2 (irfft, ky=0 once)
    int ky = i >> 7, Y = i & 127;
    int p = (ky * Y) & 127;
    float s, c; __sincosf(w0 * (float)p, &s, &c);
    float sc = ((ky == 0) ? 1.0f : 2.0f) * (1.0f / 16384.0f);
    ws[OFF_E + i] = sc * c;
    ws[OFF_E + 2048 + i] = sc * s;
  }
}

// ---------------- stage 1: per-image truncated rfft2 via two WMMA GEMMs ----
__global__ __launch_bounds__(256) void fno_fwd_dft(
    const float* __restrict__ gx, const float* __restrict__ ws,
    float* __restrict__ xfr, float* __restrict__ xfi) {
  extern __shared__ float smem[];
  float* xs  = smem;          // 16384 (phase B reuses as er|ei)
  float* ba  = smem + 16384;  // 4096
  float* tt  = smem + 20480;  // 4096  T[X][32] = (Tr | Ti)
  float* red = smem + 24576;  // 2048
  const int tid = threadIdx.x, lane = tid & 31, wave = tid >> 5;
  const int lr = lane & 15, lh = lane >> 4;
  const int img = blockIdx.x;  // n*64 + c
  const float* x = gx + (size_t)img * RSQ;

  for (int i = tid * 4; i < RSQ; i += 256 * 4)
    *(float4*)(xs + i) = *(const float4*)(x + i);
  for (int i = tid; i < 4096; i += 256) ba[i] = ws[OFF_BA + i];
  __syncthreads();

  // phase A: T[128x32] = x[128x128] * BA[128x32]
  for (int job = wave; job < 16; job += 8) {
    int m0 = (job & 7) * 16, n0 = (job >> 3) * 16;
    v8f acc = {};
    for (int k0 = 0; k0 < 128; k0 += 4) {
      v2f a, b;
      const float* ap = xs + (m0 + lr) * 128 + k0 + lh * 2;
      a.x = ap[0]; a.y = ap[1];
      b.x = ba[(k0 + lh * 2) * 32 + n0 + lr];
      b.y = ba[(k0 + lh * 2 + 1) * 32 + n0 + lr];
      acc = wmma4(a, b, acc);
    }
    for (int v = 0; v < 8; v++) tt[(m0 + lh * 8 + v) * 32 + n0 + lr] = acc[v];
  }
  __syncthreads();
  for (int i = tid; i < 8192; i += 256) xs[i] = ws[OFF_BB + i];
  __syncthreads();

  // phase B: xf[32x16] = E[32x128] * T (complex), wave = term*2 + mtile
  {
    int mtile = wave & 1, term = wave >> 1;  // 0:ErTr 1:EiTi 2:ErTi 3:EiTr
    const float* E = (term == 1 || term == 3) ? (xs + 4096) : xs;
    int tcol = (term == 1 || term == 2) ? 16 : 0;
    v8f acc = {};
    for (int k0 = 0; k0 < 128; k0 += 4) {
      v2f a, b;
      const float* ap = E + (mtile * 16 + lr) * 128 + k0 + lh * 2;
      a.x = ap[0]; a.y = ap[1];
      b.x = tt[(k0 + lh * 2) * 32 + tcol + lr];
      b.y = tt[(k0 + lh * 2 + 1) * 32 + tcol + lr];
      acc = wmma4(a, b, acc);
    }
    for (int v = 0; v < 8; v++)
      red[wave * 256 + (lh * 8 + v) * 16 + lr] = acc[v];
  }
  __syncthreads();
  for (int i = tid; i < 512; i += 256) {
    int mtile = i >> 8, idx = i & 255;
    float r  = red[(0 + mtile) * 256 + idx] - red[(2 + mtile) * 256 + idx];
    float im = red[(4 + mtile) * 256 + idx] + red[(6 + mtile) * 256 + idx];
    int mode = (mtile * 16 + (idx >> 4)) * 16 + (idx & 15);
    xfr[(size_t)mode * 4096 + img] = r;
    xfi[(size_t)mode * 4096 + img] = im;
  }
}

// ---------------- stage 2: per-mode 64x64x64 complex GEMM -------------------
__global__ __launch_bounds__(256) void fno_modemix(
    const float* __restrict__ xfr, const float* __restrict__ xfi,
    const float* __restrict__ w1r, const float* __restrict__ w1i,
    const float* __restrict__ w2r, const float* __restrict__ w2i,
    float* __restrict__ cmr, float* __restrict__ cmi) {
  extern __shared__ float smem[];
  float* sxr = smem;           // 4096 : X[n][c]
  float* sxi = smem + 4096;
  float* swr = smem + 8192;    // 4096 : W[c][o]
  float* swi = smem + 12288;
  const int tid = threadIdx.x, lane = tid & 31, wave = tid >> 5;
  const int lr = lane & 15, lh = lane >> 4;
  const int mode = blockIdx.x;          // j*16 + my
  const int j = mode >> 4, my = mode & 15;
  const float* wr = (j < 16) ? w1r : w2r;
  const float* wi = (j < 16) ? w1i : w2i;
  const int mx = (j < 16) ? j : (j - 16);
  const int woff = mx * 16 + my;
  for (int i = tid; i < 4096; i += 256) {
    sxr[i] = xfr[(size_t)mode * 4096 + i];
    sxi[i] = xfi[(size_t)mode * 4096 + i];
    swr[i] = wr[(size_t)i * 256 + woff];   // w[c][o][mx][my]
    swi[i] = wi[(size_t)i * 256 + woff];
  }
  __syncthreads();
  for (int job = wave; job < 16; job += 8) {
    int m0 = (job >> 2) * 16, n0 = (job & 3) * 16;
    v8f aRp = {}, aRm = {}, aI = {};
    for (int k0 = 0; k0 < 64; k0 += 4) {
      v2f axr, axi, bwr, bwi;
      const float* p = sxr + (m0 + lr) * 64 + k0 + lh * 2;
      axr.x = p[0]; axr.y = p[1];
      p = sxi + (m0 + lr) * 64 + k0 + lh * 2;
      axi.x = p[0]; axi.y = p[1];
      bwr.x = swr[(k0 + lh * 2) * 64 + n0 + lr];
      bwr.y = swr[(k0 + lh * 2 + 1) * 64 + n0 + lr];
      bwi.x = swi[(k0 + lh * 2) * 64 + n0 + lr];
      bwi.y = swi[(k0 + lh * 2 + 1) * 64 + n0 + lr];
      aRp = wmma4(axr, bwr, aRp);   // Xr*Wr
      aRm = wmma4(axi, bwi, aRm);   // Xi*Wi (subtract later; NEG is C-neg on f32)
      aI  = wmma4(axr, bwi, aI);    // Xr*Wi
      aI  = wmma4(axi, bwr, aI);    // + Xi*Wr
    }
    for (int v = 0; v < 8; v++) {
      int n = m0 + lh * 8 + v, o = n0 + lr;
      size_t off = ((size_t)(n * 64 + o)) * 512 + mode;
      cmr[off] = aRp[v] - aRm[v];
      cmi[off] = aI[v];
    }
  }
}

// ---------------- stage 3a: inverse transform along kx ----------------------
__global__ __launch_bounds__(256) void fno_inv_x(
    const float* __restrict__ ws,
    const float* __restrict__ cmr, const float* __restrict__ cmi,
    float* __restrict__ hr_g, float* __restrict__ hi_g) {
  extern __shared__ float smem[];
  float* fr = smem;          // 4096
  float* fi = smem + 4096;   // 4096
  float* cr = smem + 8192;   // 512
  float* ci = smem + 8704;   // 512
  const int tid = threadIdx.x, lane = tid & 31, wave = tid >> 5;
  const int lr = lane & 15, lh = lane >> 4;
  const int img = blockIdx.x;  // n*64 + o
  for (int i = tid; i < 4096; i += 256) {
    fr[i] = ws[OFF_F + i];
    fi[i] = ws[OFF_F + 4096 + i];
  }
  for (int i = tid; i < 512; i += 256) {
    cr[i] = cmr[(size_t)img * 512 + i];
    ci[i] = cmi[(size_t)img * 512 + i];
  }
  __syncthreads();
  int m0 = wave * 16;
  v8f aRp = {}, aRm = {}, aI = {};
  for (int k0 = 0; k0 < 32; k0 += 4) {
    v2f afr, afi, bcr, bci;
    const float* p = fr + (m0 + lr) * 32 + k0 + lh * 2;
    afr.x = p[0]; afr.y = p[1];
    p = fi + (m0 + lr) * 32 + k0 + lh * 2;
    afi.x = p[0]; afi.y = p[1];
    bcr.x = cr[(k0 + lh * 2) * 16 + lr];
    bcr.y = cr[(k0 + lh * 2 + 1) * 16 + lr];
    bci.x = ci[(k0 + lh * 2) * 16 + lr];
    bci.y = ci[(k0 + lh * 2 + 1) * 16 + lr];
    aRp = wmma4(afr, bcr, aRp);
    aRm = wmma4(afi, bci, aRm);
    aI  = wmma4(afr, bci, aI);
    aI  = wmma4(afi, bcr, aI);
  }
  for (int v = 0; v < 8; v++) {
    int X = m0 + lh * 8 + v;
    size_t off = (size_t)img * 2048 + X * 16 + lr;
    hr_g[off] = aRp[v] - aRm[v];
    hi_g[off] = aI[v];
  }
}

// ------- stage 3b + residual + subdomain gather (memory-bound, VALU) --------
__global__ __launch_bounds__(256) void fno_gather_out(
    const float* __restrict__ gx, const int* __restrict__ lookup,
    const float* __restrict__ ws,
    const float* __restrict__ hr_g, const float* __restrict__ hi_g,
    float* __restrict__ out) {
  __shared__ float e[4096];
  const int tid = threadIdx.x;
  for (int i = tid; i < 4096; i += 256) e[i] = ws[OFF_E + i];
  __syncthreads();
  int g = blockIdx.x * 256 + tid;
  int pix = g & 16383;
  int bc = g >> 14;
  int b = bc >> 6, c = bc & 63;
  int s = lookup[pix];
  int im = (b * 16 + s) * 64 + c;
  float acc = gx[(size_t)im * RSQ + pix];     // residual
  int Y = pix & 127;
  const float* hr = hr_g + (size_t)im * 2048 + (pix >> 7) * 16;
  const float* hi = hi_g + (size_t)im * 2048 + (pix >> 7) * 16;
#pragma unroll
  for (int ky = 0; ky < 16; ky++)
    acc += hr[ky] * e[ky * 128 + Y] - hi[ky] * e[2048 + ky * 128 + Y];
  out[g] = acc;
}

extern "C" void kernel_launch(void* const* d_in, const int* in_sizes, int n_in,
                              void* d_out, int out_size, void* d_ws,
                              size_t ws_size, hipStream_t stream) {
  (void)in_sizes; (void)n_in; (void)out_size; (void)ws_size;
  const float* ref    = (const float*)d_in[0];
  const int*   lookup = (const int*)d_in[2];
  const float* w1r    = (const float*)d_in[3];
  const float* w1i    = (const float*)d_in[4];
  const float* w2r    = (const float*)d_in[5];
  const float* w2i    = (const float*)d_in[6];
  float* ws  = (float*)d_ws;   // requires ~101 MB of scratch
  float* out = (float*)d_out;

  fno_init_basis<<<16, 256, 0, stream>>>(ws);
  fno_fwd_dft<<<NI * 64, 256, 26624 * sizeof(float), stream>>>(
      ref, ws, ws + OFF_XFR, ws + OFF_XFI);
  fno_modemix<<<512, 256, 16384 * sizeof(float), stream>>>(
      ws + OFF_XFR, ws + OFF_XFI, w1r, w1i, w2r, w2i,
      ws + OFF_CMR, ws + OFF_CMI);
  fno_inv_x<<<NI * 64, 256, 9216 * sizeof(float), stream>>>(
      ws, ws + OFF_CMR, ws + OFF_CMI, ws + OFF_HR, ws + OFF_HI);
  fno_gather_out<<<16384, 256, 0, stream>>>(
      ref, lookup, ws, ws + OFF_HR, ws + OFF_HI, out);
}